// MinDistanceLoss_41094247088549
// MI455X (gfx1250) — compile-verified
//
#include <hip/hip_runtime.h>
#include <hip/hip_bf16.h>

typedef __attribute__((ext_vector_type(2))) float v2f;
typedef __attribute__((ext_vector_type(8))) float v8f;

#define ATOMS 64   // atoms per structure (setup: A = 64, uniform)

// One block (64 threads = 2 waves) per structure.
// Phase 1: per-atom w = M^T(M(xt-x)), per-structure q(o)=|M o|^2 (27 values).
// Phase 2: score[64x32] = q(o) - 2 w.o  via v_wmma_f32_16x16x4_f32
//          (A = w  [16x4 f32 layout], B = -2*OFFSETS chunk [4x16], C = q broadcast).
// Phase 3: per-atom argmin(27) -> shifted; fixed-order segment reductions -> partial loss.
__global__ __launch_bounds__(64) void mdl_structure_kernel(
    const float* __restrict__ cell,       // [B,3,3]
    const float* __restrict__ x,          // [N,3]
    const float* __restrict__ xt,         // [N,3]
    const int*   __restrict__ num_atoms,  // [B]
    float* __restrict__ partial)          // [B] per-structure sum of |xt - pred|
{
  __shared__ float sM[9];                 // cell, row-major
  __shared__ float sW[ATOMS][4];          // w vectors (padded stride 4)
  __shared__ float sQ[32];                // q(o), padded 27 -> 32
  __shared__ float sScore[ATOMS][32];     // WMMA output scores
  __shared__ float sRed[ATOMS][4];        // reduction staging
  __shared__ float sCenter[3];

  const int tid  = threadIdx.x;
  const int lane = tid & 31;
  const int wv   = tid >> 5;
  const int b    = blockIdx.x;
  const int n    = b * ATOMS + tid;

  if (tid < 9) sM[tid] = cell[b * 9 + tid];
  __syncthreads();

  const float m00 = sM[0], m01 = sM[1], m02 = sM[2];
  const float m10 = sM[3], m11 = sM[4], m12 = sM[5];
  const float m20 = sM[6], m21 = sM[7], m22 = sM[8];

  const float x0 = x[n * 3 + 0], x1 = x[n * 3 + 1], x2 = x[n * 3 + 2];
  const float t0 = xt[n * 3 + 0], t1 = xt[n * 3 + 1], t2 = xt[n * 3 + 2];
  const float d0 = t0 - x0, d1 = t1 - x1, d2 = t2 - x2;

  // v = M d ; w = M^T v  (so score(o) = |M o|^2 - 2 w.o + const)
  const float v0 = m00 * d0 + m01 * d1 + m02 * d2;
  const float v1 = m10 * d0 + m11 * d1 + m12 * d2;
  const float v2 = m20 * d0 + m21 * d1 + m22 * d2;
  sW[tid][0] = m00 * v0 + m10 * v1 + m20 * v2;
  sW[tid][1] = m01 * v0 + m11 * v1 + m21 * v2;
  sW[tid][2] = m02 * v0 + m12 * v1 + m22 * v2;
  sW[tid][3] = 0.f;

  if (tid < 32) {
    float qv = 3.0e38f;                    // padded offsets can never win
    if (tid < 27) {
      const float o0 = (float)(tid / 9) - 1.f;
      const float o1 = (float)((tid / 3) % 3) - 1.f;
      const float o2 = (float)(tid % 3) - 1.f;
      const float g0 = m00 * o0 + m01 * o1 + m02 * o2;
      const float g1 = m10 * o0 + m11 * o1 + m12 * o2;
      const float g2 = m20 * o0 + m21 * o1 + m22 * o2;
      qv = g0 * g0 + g1 * g1 + g2 * g2;
    }
    sQ[tid] = qv;
  }
  __syncthreads();

  // ---- WMMA scoring: wave wv covers atom groups {2wv, 2wv+1}, each x 2 offset chunks
#pragma unroll
  for (int gi = 0; gi < 2; ++gi) {
    const int g = wv * 2 + gi;
    const int m = g * 16 + (lane & 15);
    // A (16x4 f32): lanes 0-15 -> K=0,1 ; lanes 16-31 -> K=2,3 (K=3 is zero pad)
    v2f a;
    if (lane < 16) { a.x = sW[m][0]; a.y = sW[m][1]; }
    else           { a.x = sW[m][2]; a.y = 0.f; }
#pragma unroll
    for (int c = 0; c < 2; ++c) {
      const int off = c * 16 + (lane & 15);
      float o0 = 0.f, o1 = 0.f, o2 = 0.f;
      if (off < 27) {
        o0 = (float)(off / 9) - 1.f;
        o1 = (float)((off / 3) % 3) - 1.f;
        o2 = (float)(off % 3) - 1.f;
      }
      // B (4x16 f32), mirrored layout: lanes 0-15 -> K=0,1 rows ; lanes 16-31 -> K=2,3
      v2f bb;
      if (lane < 16) { bb.x = -2.f * o0; bb.y = -2.f * o1; }
      else           { bb.x = -2.f * o2; bb.y = 0.f; }
      // C = q(o) broadcast down each column (same value in all 8 row-VGPRs)
      const float qv = sQ[off];
      v8f cm;
#pragma unroll
      for (int r = 0; r < 8; ++r) cm[r] = qv;

      v8f dmat = __builtin_amdgcn_wmma_f32_16x16x4_f32(
          /*neg_a=*/false, a, /*neg_b=*/false, bb,
          /*c_mod=*/(short)0, cm, /*reuse_a=*/false, /*reuse_b=*/false);

      const int half = (lane < 16) ? 0 : 8;   // D: VGPR r holds rows M=r / M=r+8
#pragma unroll
      for (int r = 0; r < 8; ++r)
        sScore[g * 16 + r + half][c * 16 + (lane & 15)] = dmat[r];
    }
  }
  __syncthreads();

  // ---- argmin over 27 offsets, first-min tie-break (matches jnp.argmin)
  float best = sScore[tid][0];
  int   bi = 0;
#pragma unroll
  for (int o = 1; o < 27; ++o) {
    const float s = sScore[tid][o];
    if (s < best) { best = s; bi = o; }
  }
  const float s0 = x0 + ((float)(bi / 9) - 1.f);
  const float s1 = x1 + ((float)((bi / 3) % 3) - 1.f);
  const float s2 = x2 + ((float)(bi % 3) - 1.f);

  // ---- center = segment_mean(shifted - xt): fixed-order serial sums (deterministic)
  sRed[tid][0] = s0 - t0;
  sRed[tid][1] = s1 - t1;
  sRed[tid][2] = s2 - t2;
  __syncthreads();
  if (tid < 3) {
    float acc = 0.f;
    for (int i = 0; i < ATOMS; ++i) acc += sRed[i][tid];
    sCenter[tid] = acc;
  }
  __syncthreads();

  const float inv_na = 1.f / (float)num_atoms[b];
  const float c0 = sCenter[0] * inv_na;
  const float c1 = sCenter[1] * inv_na;
  const float c2 = sCenter[2] * inv_na;
  const float p0 = s0 - c0, p1 = s1 - c1, p2 = s2 - c2;

  sRed[tid][0] = fabsf(t0 - p0) + fabsf(t1 - p1) + fabsf(t2 - p2);
  __syncthreads();
  if (tid == 0) {
    float acc = 0.f;
    for (int i = 0; i < ATOMS; ++i) acc += sRed[i][0];
    partial[b] = acc;
  }
}

// Deterministic final reduction: fixed strided accumulation + fixed binary tree.
__global__ __launch_bounds__(256) void mdl_reduce_kernel(
    const float* __restrict__ partial, int nB, float inv_total,
    float* __restrict__ out)
{
  __shared__ float s[256];
  const int tid = threadIdx.x;
  float acc = 0.f;
  for (int i = tid; i < nB; i += 256) acc += partial[i];
  s[tid] = acc;
  __syncthreads();
  for (int st = 128; st > 0; st >>= 1) {
    if (tid < st) s[tid] += s[tid + st];
    __syncthreads();
  }
  if (tid == 0) out[0] = s[0] * inv_total;
}

extern "C" void kernel_launch(void* const* d_in, const int* in_sizes, int n_in,
                              void* d_out, int out_size, void* d_ws, size_t ws_size,
                              hipStream_t stream) {
  const float* cell = (const float*)d_in[0];  // [B,3,3]
  const float* x    = (const float*)d_in[1];  // [N,3]
  const float* xt   = (const float*)d_in[2];  // [N,3]
  const int*   na   = (const int*)d_in[3];    // [B]

  const int B = in_sizes[0] / 9;
  const int N = in_sizes[1] / 3;

  float* partial = (float*)d_ws;              // B floats (64 KB) of scratch

  mdl_structure_kernel<<<B, ATOMS, 0, stream>>>(cell, x, xt, na, partial);

  const float inv_total = 1.f / (3.f * (float)N);
  mdl_reduce_kernel<<<1, 256, 0, stream>>>(partial, B, inv_total, (float*)d_out);
}